// AttnDecoderRNN_55336358641946
// MI455X (gfx1250) — compile-verified
//
#include <hip/hip_runtime.h>
#include <hip/hip_bf16.h>
#include <math.h>

// ---------------------------------------------------------------------------
// AttnDecoderRNN step, batch=1.  H=1024, V=50257, S=8192.
// Entirely bandwidth bound (~515 MB traffic, ~22us at 23.3 TB/s).
// All GEMVs run through V_WMMA_F32_16X16X4_F32: one wave = 16 output rows.
// Key algebraic rewrite: (enc @ W_attn.T + b) @ h  ==  enc @ (W_attn.T h) + b.h
// (8.6 GMAC -> 9 MMAC, removes ~34 GB of effective traffic).
// ---------------------------------------------------------------------------

typedef __attribute__((ext_vector_type(2))) float v2f;
typedef __attribute__((ext_vector_type(8))) float v8f;

#define HDIM 1024
#define VDIM 50257
#define SDIM 8192

// ---------------------------------------------------------------------------
// GEMV via WMMA: y[r] = sum_k W[r][k]*x[k] (+bias[r]) (+*scalar_add)
// One wave handles 16 rows.  Lane l: m = l&15 (row), kh = 2*(l>>4).
// A tile (16x4 f32):  lane loads float2 of its row at col k+kh   -> exact
//   ISA layout (VGPR0 = K=kh, VGPR1 = K=kh+1 per lane group).
// B tile (4x16 f32):  lane loads the SAME float2 of the x vector -> every
//   N column of B equals x[k..k+3], so all 16 columns of D hold the same
//   dot product; lanes 0 / 16 read rows 0..7 / 8..15 out of the 8 acc VGPRs.
// NT template: non-temporal weight streaming (W_out is 412 MB single-use and
// must not evict enc, which is re-read and fits in the 192 MB L2).
// ---------------------------------------------------------------------------
template <bool NT>
__global__ void gemv16_wmma(const float* __restrict__ W,
                            const float* __restrict__ x,
                            const float* __restrict__ bias,
                            const float* __restrict__ scalar_add,
                            const int* __restrict__ idx,
                            float* __restrict__ y,
                            int R, int K) {
  const int lane = threadIdx.x & 31;
  const int wave = threadIdx.x >> 5;
  const int tile = blockIdx.x * (blockDim.x >> 5) + wave;
  const int row_base = tile * 16;
  if (row_base >= R) return;                 // wave-uniform exit, EXEC stays full

  const int m  = lane & 15;
  const int kh = (lane >> 4) << 1;           // 0 or 2
  int row = row_base + m;
  if (row > R - 1) row = R - 1;              // clamp OOB loads (store is guarded)

  const float* __restrict__ xv = idx ? (x + (long)idx[0] * K) : x;
  const float* __restrict__ wr = W + (long)row * K + kh;
  const float* __restrict__ xp = xv + kh;

  v8f acc = {};
#pragma unroll 4
  for (int k = 0; k < K; k += 4) {
    v2f a = NT ? __builtin_nontemporal_load((const v2f*)(wr + k))
               : *(const v2f*)(wr + k);
    v2f b = *(const v2f*)(xp + k);           // wave-uniform, cache-resident
    acc = __builtin_amdgcn_wmma_f32_16x16x4_f32(
        /*neg_a=*/false, a, /*neg_b=*/false, b,
        /*c_mod=*/(short)0, acc, /*reuse_a=*/false, /*reuse_b=*/false);
  }

  const float s = scalar_add ? scalar_add[0] : 0.0f;
  // D layout: VGPR j, lanes 0-15 -> M=j ; lanes 16-31 -> M=j+8 (all N equal)
  if (lane == 0 || lane == 16) {
    const int rb = row_base + ((lane >> 4) << 3);
#pragma unroll
    for (int j = 0; j < 8; ++j) {
      const int r = rb + j;
      if (r < R) y[r] = acc[j] + (bias ? bias[r] : 0.0f) + s;
    }
  }
}

// ---------------------------------------------------------------------------
// GRU gate fusion: r,z,n -> h_new (written to ws and to d_out h_new slot)
// ---------------------------------------------------------------------------
__global__ void gru_gate(const float* __restrict__ gi, const float* __restrict__ gh,
                         const float* __restrict__ h,
                         float* __restrict__ hn0, float* __restrict__ hn1, int H) {
  const int t = blockIdx.x * blockDim.x + threadIdx.x;
  if (t >= H) return;
  const float ir = gi[t], iz = gi[H + t], in_ = gi[2 * H + t];
  const float hr = gh[t], hz = gh[H + t], hn = gh[2 * H + t];
  const float r = 1.0f / (1.0f + expf(-(ir + hr)));
  const float z = 1.0f / (1.0f + expf(-(iz + hz)));
  const float n = tanhf(in_ + r * hn);
  const float hv = h[t];
  const float hnew = (1.0f - z) * n + z * hv;
  hn0[t] = hnew;
  hn1[t] = hnew;
}

// ---------------------------------------------------------------------------
// v[j] += sum_{i in chunk} W_attn[i][j] * h[i]   (transposed GEMV, coalesced
// across columns, f32 global atomics to accumulate row-chunks)
// ---------------------------------------------------------------------------
__global__ void attn_v_partial(const float* __restrict__ Wt, const float* __restrict__ h,
                               float* __restrict__ v, int H, int ROWS) {
  const int j  = blockIdx.x * blockDim.x + threadIdx.x;
  const int i0 = blockIdx.y * ROWS;
  float acc = 0.0f;
  for (int i = 0; i < ROWS; ++i)
    acc += Wt[(long)(i0 + i) * H + j] * h[i0 + i];
  atomicAdd(&v[j], acc);
}

// one-block dot product (b_attn . h_new) -> scalar
__global__ void dot_reduce(const float* __restrict__ a, const float* __restrict__ b,
                           float* __restrict__ out, int n) {
  __shared__ float red[256];
  float acc = 0.0f;
  for (int i = threadIdx.x; i < n; i += 256) acc += a[i] * b[i];
  red[threadIdx.x] = acc;
  __syncthreads();
  for (int s = 128; s > 0; s >>= 1) {
    if (threadIdx.x < s) red[threadIdx.x] += red[threadIdx.x + s];
    __syncthreads();
  }
  if (threadIdx.x == 0) out[0] = red[0];
}

// one-block softmax over S scores (S*4 = 32 KB, trivially cache resident)
__global__ void softmax_S(const float* __restrict__ scores, float* __restrict__ attn, int S) {
  __shared__ float red[256];
  const int t = threadIdx.x;
  float m = -INFINITY;
  for (int i = t; i < S; i += 256) m = fmaxf(m, scores[i]);
  red[t] = m;
  __syncthreads();
  for (int s = 128; s > 0; s >>= 1) {
    if (t < s) red[t] = fmaxf(red[t], red[t + s]);
    __syncthreads();
  }
  m = red[0];
  __syncthreads();
  float sum = 0.0f;
  for (int i = t; i < S; i += 256) sum += expf(scores[i] - m);
  red[t] = sum;
  __syncthreads();
  for (int s = 128; s > 0; s >>= 1) {
    if (t < s) red[t] += red[t + s];
    __syncthreads();
  }
  const float inv = 1.0f / red[0];
  for (int i = t; i < S; i += 256) attn[i] = expf(scores[i] - m) * inv;
}

// ctx[col] += sum_{s in chunk} attn[s] * enc[s][col]  (coalesced, f32 atomics)
__global__ void context_partial(const float* __restrict__ enc, const float* __restrict__ attn,
                                float* __restrict__ ctx, int H, int SROWS) {
  const int col = blockIdx.x * blockDim.x + threadIdx.x;
  const int s0  = blockIdx.y * SROWS;
  float acc = 0.0f;
  for (int s = 0; s < SROWS; ++s)
    acc += attn[s0 + s] * enc[(long)(s0 + s) * H + col];
  atomicAdd(&ctx[col], acc);
}

// ---------------------------------------------------------------------------
extern "C" void kernel_launch(void* const* d_in, const int* in_sizes, int n_in,
                              void* d_out, int out_size, void* d_ws, size_t ws_size,
                              hipStream_t stream) {
  (void)in_sizes; (void)n_in; (void)out_size; (void)ws_size;

  const int*   word  = (const int*)  d_in[0];
  const float* hlast = (const float*)d_in[1];   // (1,1,H)
  const float* enc   = (const float*)d_in[2];   // (S,1,H) -> (S,H)
  const float* emb   = (const float*)d_in[3];   // (V,H)
  const float* Wat   = (const float*)d_in[4];   // (H,H)
  const float* bat   = (const float*)d_in[5];   // (H,)
  const float* Wih   = (const float*)d_in[6];   // (3H,H)
  const float* bih   = (const float*)d_in[7];
  const float* Whh   = (const float*)d_in[8];   // (3H,H)
  const float* bhh   = (const float*)d_in[9];
  const float* Wout  = (const float*)d_in[10];  // (V,2H)
  const float* bout  = (const float*)d_in[11];

  float* out_logits = (float*)d_out;                  // [V]
  float* out_hnew   = out_logits + VDIM;              // [H]
  float* out_attn   = out_hnew + HDIM;                // [S]

  // workspace layout (floats); h_new and context are ADJACENT -> the final
  // GEMV reads concat([h_new, context]) as one contiguous 2H vector.
  float* ws       = (float*)d_ws;
  float* ws_gi    = ws;                 // 3072
  float* ws_gh    = ws + 3072;          // 3072
  float* ws_hnew  = ws + 6144;          // 1024  \ contiguous 2H
  float* ws_ctx   = ws + 7168;          // 1024  /
  float* ws_v     = ws + 8192;          // 1024
  float* ws_c     = ws + 9216;          // 1
  float* ws_score = ws + 9220;          // 8192

  // zero the atomically-accumulated regions (ctx, v, c) every call
  hipMemsetAsync((char*)d_ws + 7168 * sizeof(float), 0, 2049 * sizeof(float), stream);

  // gi = W_ih @ emb[word] + b_ih ;  gh = W_hh @ h + b_hh   (3072 rows, K=1024)
  gemv16_wmma<true><<<24, 256, 0, stream>>>(Wih, emb,   bih, nullptr, word,    ws_gi, 3 * HDIM, HDIM);
  gemv16_wmma<true><<<24, 256, 0, stream>>>(Whh, hlast, bhh, nullptr, nullptr, ws_gh, 3 * HDIM, HDIM);

  // GRU gates -> h_new (to ws and to d_out)
  gru_gate<<<4, 256, 0, stream>>>(ws_gi, ws_gh, hlast, ws_hnew, out_hnew, HDIM);

  // v = W_attn^T @ h_new  ;  c = b_attn . h_new
  attn_v_partial<<<dim3(4, 8), 256, 0, stream>>>(Wat, ws_hnew, ws_v, HDIM, 128);
  dot_reduce<<<1, 256, 0, stream>>>(bat, ws_hnew, ws_c, HDIM);

  // scores = enc @ v + c   (8192 rows, K=1024) — keep enc temporal (re-read below)
  gemv16_wmma<false><<<64, 256, 0, stream>>>(enc, ws_v, nullptr, ws_c, nullptr, ws_score, SDIM, HDIM);

  // attn = softmax(scores) -> d_out
  softmax_S<<<1, 256, 0, stream>>>(ws_score, out_attn, SDIM);

  // context = attn @ enc
  context_partial<<<dim3(4, 16), 256, 0, stream>>>(enc, out_attn, ws_ctx, HDIM, 512);

  // logits = W_out @ [h_new; context] + b_out   (50257 rows, K=2048, 412 MB NT)
  gemv16_wmma<true><<<393, 256, 0, stream>>>(Wout, ws_hnew, bout, nullptr, nullptr, out_logits, VDIM, 2 * HDIM);
}